// GNNEncoder_76209899701045
// MI455X (gfx1250) — compile-verified
//
#include <hip/hip_runtime.h>

#define D 64
#define LDSW 68            // padded LDS row stride (floats) -> conflict-free ds_load_b64
#define BN_EPS 1e-5f

typedef __attribute__((ext_vector_type(2))) float v2f;
typedef __attribute__((ext_vector_type(8))) float v8f;

// ---------------------------------------------------------------------------
// Zero the scatter accumulators (s followed contiguously by cnt).
// ---------------------------------------------------------------------------
__global__ void zero_f32_kernel(float* __restrict__ p, int n) {
    int i = blockIdx.x * blockDim.x + threadIdx.x;
    if (i < n) p[i] = 0.0f;
}

// ---------------------------------------------------------------------------
// Gather x[src] and scatter-add into s[dst]; count edges per dst.
// One thread handles one edge x 16-byte feature chunk (float4).
// x (25.6MB) is L2-resident, so this is L2-bandwidth / atomic bound.
// ---------------------------------------------------------------------------
__global__ void scatter_mean_kernel(const float* __restrict__ x,
                                    const int*   __restrict__ ei,
                                    float* __restrict__ s,
                                    float* __restrict__ cnt,
                                    int n_edges) {
    int idx = blockIdx.x * blockDim.x + threadIdx.x;
    int e = idx >> 4;
    if (e >= n_edges) return;
    int c   = idx & 15;
    int src = ei[e];
    int dst = ei[n_edges + e];
    const float4 v = *(const float4*)(x + (size_t)src * D + c * 4);
    float* sp = s + (size_t)dst * D + c * 4;
    atomicAdd(sp + 0, v.x);
    atomicAdd(sp + 1, v.y);
    atomicAdd(sp + 2, v.z);
    atomicAdd(sp + 3, v.w);
    if (c == 0) atomicAdd(cnt + dst, 1.0f);
}

// ---------------------------------------------------------------------------
// Fused per-node epilogue: agg = s/max(cnt,1);
// out = BN(ELU(agg @ Wl^T + b + x @ Wr^T))
// One wave32 per 16-node tile; full-precision V_WMMA_F32_16X16X4_F32.
//
// A-fragment layout (16x4 f32): lane m=lane&15 holds rows, VGPR pair holds
// K = kt*4 + (lane>>4)*2 + {0,1}.
// B-fragment (4x16 = Wl^T tile): lane supplies column j = jt*16 + (lane&15),
// pair of K values from LDS (row stride 68 -> all 64 banks hit once).
// C/D layout: lane holds column j, rows r + 8*(lane>>4), r = VGPR 0..7.
// ---------------------------------------------------------------------------
__global__ void __launch_bounds__(256)
sage_fused_kernel(const float* __restrict__ xin,
                  const float* __restrict__ s,
                  const float* __restrict__ cnt,
                  const float* __restrict__ wl,
                  const float* __restrict__ bl,
                  const float* __restrict__ wr,
                  const float* __restrict__ bng,
                  const float* __restrict__ bnb,
                  const float* __restrict__ bnm,
                  const float* __restrict__ bnv,
                  float* __restrict__ out,
                  int n_nodes) {
    __shared__ float lwl[D * LDSW];
    __shared__ float lwr[D * LDSW];

    int t = threadIdx.x;
    // Cooperative stage of both weight matrices into LDS (padded rows).
    for (int i = t; i < D * (D / 4); i += 256) {
        int row = i >> 4;
        int c4  = i & 15;
        *(float4*)(lwl + row * LDSW + c4 * 4) = *(const float4*)(wl + row * D + c4 * 4);
        *(float4*)(lwr + row * LDSW + c4 * 4) = *(const float4*)(wr + row * D + c4 * 4);
    }
    __syncthreads();

    int wave  = t >> 5;          // 8 waves / block (wave32)
    int lane  = t & 31;
    int node0 = (blockIdx.x * 8 + wave) * 16;
    if (node0 >= n_nodes) return;   // wave-uniform: EXEC stays all-ones for WMMA

    int m  = lane & 15;          // row (A) / column (B,C,D) index within tile
    int kh = lane >> 4;          // K-half select
    int node = node0 + m;

    float rinv = 1.0f / fmaxf(cnt[node], 1.0f);

    // Build all A-fragments for this 16-node tile (K = 64 -> 16 tiles of 4).
    v2f a_agg[16], a_x[16];
    const float* srow = s   + (size_t)node * D + kh * 2;
    const float* xrow = xin + (size_t)node * D + kh * 2;
#pragma unroll
    for (int kt = 0; kt < 16; ++kt) {
        v2f sv = *(const v2f*)(srow + kt * 4);
        v2f xv = *(const v2f*)(xrow + kt * 4);
        a_agg[kt] = sv * rinv;   // fold mean-division into A fragment
        a_x[kt]   = xv;
    }

#pragma unroll
    for (int jt = 0; jt < 4; ++jt) {
        int j = jt * 16 + m;
        const float* bcolL = lwl + j * LDSW + kh * 2;
        const float* bcolR = lwr + j * LDSW + kh * 2;

        v8f acc = {};
#pragma unroll
        for (int kt = 0; kt < 16; ++kt) {
            v2f bf = *(const v2f*)(bcolL + kt * 4);
            acc = __builtin_amdgcn_wmma_f32_16x16x4_f32(
                false, a_agg[kt], false, bf, (short)0, acc, false, false);
        }
#pragma unroll
        for (int kt = 0; kt < 16; ++kt) {
            v2f bf = *(const v2f*)(bcolR + kt * 4);
            acc = __builtin_amdgcn_wmma_f32_16x16x4_f32(
                false, a_x[kt], false, bf, (short)0, acc, false, false);
        }

        // Per-column epilogue params (column == lane&15, constant over rows).
        float bj    = bl[j];
        float scale = bng[j] * rsqrtf(bnv[j] + BN_EPS);
        float shift = bnb[j] - bnm[j] * scale;

        float* op = out + (size_t)(node0 + kh * 8) * D + j;
#pragma unroll
        for (int r = 0; r < 8; ++r) {
            float v = acc[r] + bj;
            v = (v > 0.0f) ? v : (__expf(v) - 1.0f);   // ELU
            op[(size_t)r * D] = v * scale + shift;     // eval BatchNorm
        }
    }
}

// ---------------------------------------------------------------------------
extern "C" void kernel_launch(void* const* d_in, const int* in_sizes, int n_in,
                              void* d_out, int out_size, void* d_ws, size_t ws_size,
                              hipStream_t stream) {
    const float* x   = (const float*)d_in[0];
    const int*   ei  = (const int*)  d_in[1];
    const float* w1l = (const float*)d_in[2];
    const float* b1l = (const float*)d_in[3];
    const float* w1r = (const float*)d_in[4];
    const float* w2l = (const float*)d_in[5];
    const float* b2l = (const float*)d_in[6];
    const float* w2r = (const float*)d_in[7];
    const float* g1  = (const float*)d_in[8];
    const float* be1 = (const float*)d_in[9];
    const float* m1  = (const float*)d_in[10];
    const float* v1  = (const float*)d_in[11];
    const float* g2  = (const float*)d_in[12];
    const float* be2 = (const float*)d_in[13];
    const float* m2  = (const float*)d_in[14];
    const float* v2  = (const float*)d_in[15];

    const int n_nodes = in_sizes[0] / D;
    const int n_edges = in_sizes[1] / 2;

    // Workspace layout (contiguous): s [N*D] | cnt [N] | h1 [N*D]
    float* s   = (float*)d_ws;
    float* cnt = s + (size_t)n_nodes * D;
    float* h1  = cnt + n_nodes;
    float* out = (float*)d_out;

    const int zn = n_nodes * D + n_nodes;
    dim3 zgrid((zn + 255) / 256);
    dim3 sgrid((unsigned)(((size_t)n_edges * 16 + 255) / 256));
    dim3 fgrid((n_nodes + 127) / 128);   // 8 waves x 16 nodes per block

    // ---- Layer 1: SAGEConv -> ELU -> BN ----
    zero_f32_kernel<<<zgrid, 256, 0, stream>>>(s, zn);
    scatter_mean_kernel<<<sgrid, 256, 0, stream>>>(x, ei, s, cnt, n_edges);
    sage_fused_kernel<<<fgrid, 256, 0, stream>>>(x, s, cnt, w1l, b1l, w1r,
                                                 g1, be1, m1, v1, h1, n_nodes);

    // ---- Layer 2: SAGEConv -> ELU -> BN ----
    zero_f32_kernel<<<zgrid, 256, 0, stream>>>(s, zn);
    scatter_mean_kernel<<<sgrid, 256, 0, stream>>>(h1, ei, s, cnt, n_edges);
    sage_fused_kernel<<<fgrid, 256, 0, stream>>>(h1, s, cnt, w2l, b2l, w2r,
                                                 g2, be2, m2, v2, out, n_nodes);
}